// Model_54211077210330
// MI455X (gfx1250) — compile-verified
//
#include <hip/hip_runtime.h>

typedef float v2f __attribute__((ext_vector_type(2)));
typedef float v8f __attribute__((ext_vector_type(8)));

// ---------------------------------------------------------------------------
// Generic batched WMMA GEMM:  C[M,N] = A[M,K] @ B[K,N]  (TRANSB: B is [N,K]).
// f32 via V_WMMA_F32_16X16X4_F32. Requirements (guaranteed by all call sites):
//   - M multiple of 16 (no A-row masking)
//   - B buffer padded so any column in a live 16-wide tile is loadable
//     (pad columns feed only discarded output columns)
//   - K multiple of 4, lda/ldb even (8-byte aligned v2f loads)
// 8 waves / 256-thread block, each wave owns one 16x16 C tile.
// Epilogue modes:
//   0: none
//   1: (acc + p0[col]) * p1[col] + p2[col]            (linear + BN affine)
//   2: acc + p0[col]                                  (bias)
//   3: leaky(acc + p0[col])                           (bias + leaky relu)
//   4: leaky(acc - (row==col)*1e8)                    (adjacency mask)
//   5: relu(acc * p0[row] + p1[row])                  (conv BN-as-scale + relu)
// ---------------------------------------------------------------------------
template <int TRANSB>
__global__ __launch_bounds__(256) void k_gemm_wmma_f32(
    const float* __restrict__ A, const float* __restrict__ B, float* __restrict__ C,
    int M, int N, int K, int lda, int ldb, int ldc,
    long long sA, long long sB, long long sC,
    int mode,
    const float* __restrict__ p0, const float* __restrict__ p1,
    const float* __restrict__ p2)
{
    const int wave = threadIdx.x >> 5;
    const int lane = threadIdx.x & 31;
    const int hf   = lane >> 4;     // half-wave: selects K pair / M-row group
    const int lm   = lane & 15;
    const int m0   = blockIdx.x * 16;
    const int nt   = blockIdx.y * 8 + wave;
    const int n0   = nt * 16;
    if (n0 >= N) return;            // wave-uniform: EXEC stays all-ones for WMMA
    const long long bz = blockIdx.z;
    A += bz * sA; B += bz * sB; C += bz * sC;

    const int arow = m0 + lm;
    const int bcol = n0 + lm;
    const float* Ap = A + (size_t)arow * lda;
    const float* Bp = TRANSB ? (B + (size_t)bcol * ldb) : (B + bcol);

    v8f acc = {};
    #pragma unroll 4
    for (int k0 = 0; k0 < K; k0 += 4) {
        const int ka = k0 + (hf << 1);  // lanes 0-15: k0,k0+1 ; lanes 16-31: k0+2,k0+3
        v2f af = *(const v2f*)(Ap + ka);
        v2f bf;
        if (TRANSB) {
            bf = *(const v2f*)(Bp + ka);
        } else {
            bf.x = Bp[(size_t)ka * ldb];
            bf.y = Bp[(size_t)ka * ldb + ldb];
        }
        acc = __builtin_amdgcn_wmma_f32_16x16x4_f32(
            false, af, false, bf, (short)0, acc, false, false);
    }

    if (bcol < N) {
        #pragma unroll
        for (int v = 0; v < 8; ++v) {
            int r = m0 + v + (hf << 3);   // D VGPR v: rows v (lanes 0-15) / v+8 (16-31)
            float val = acc[v];
            if (mode == 1)      { val = (val + p0[bcol]) * p1[bcol] + p2[bcol]; }
            else if (mode == 2) { val = val + p0[bcol]; }
            else if (mode == 3) { val += p0[bcol]; val = val < 0.f ? val * 0.01f : val; }
            else if (mode == 4) { if (r == bcol) val -= 1e8f;
                                  val = val < 0.f ? val * 0.01f : val; }
            else if (mode == 5) { val = val * p0[r] + p1[r]; val = fmaxf(val, 0.f); }
            C[(size_t)r * ldc + bcol] = val;
        }
    }
}

// im2col for kernel=8, pad=4, stride=1, writing a column-padded buffer:
// out[b][i*8+k][t] = in[b][i][t+k-4] (0 if OOB), for t in [0, Ltp)
__global__ void k_im2col_k8p4(const float* __restrict__ in, float* __restrict__ out,
                              int I, int Lin, int Ltp,
                              long long inStride, long long outStride)
{
    int tot = I * 8 * Ltp;
    int idx = blockIdx.x * blockDim.x + threadIdx.x;
    if (idx >= tot) return;
    long long b = blockIdx.z;
    int t = idx % Ltp;
    int r = idx / Ltp;
    int i = r >> 3;
    int k = r & 7;
    int src = t + k - 4;
    float v = 0.f;
    if (src >= 0 && src < Lin) v = in[b * inStride + (size_t)i * Lin + src];
    out[b * outStride + (size_t)r * Ltp + t] = v;
}

// MaxPool1d(kernel=2, stride=2, pad=1) with -inf padding (torch semantics):
// out[c][u] = max(in[c][2u-1], in[c][2u]) with OOB treated as -inf.
__global__ void k_pool2p1(const float* __restrict__ in, float* __restrict__ out,
                          int C, int Lin, int Lout,
                          long long inStride, long long outStride)
{
    int tot = C * Lout;
    int idx = blockIdx.x * blockDim.x + threadIdx.x;
    if (idx >= tot) return;
    long long b = blockIdx.z;
    int u = idx % Lout;
    int c = idx / Lout;
    const float* p = in + b * inStride + (size_t)c * Lin;
    int lo = 2 * u - 1, hi = 2 * u;
    float v = -__builtin_inff();
    if (lo >= 0 && lo < Lin) v = p[lo];
    if (hi >= 0 && hi < Lin) v = fmaxf(v, p[hi]);
    out[b * outStride + (size_t)c * Lout + u] = v;
}

// Positional-encoding + the reshape/transpose dance, collapsed to index math:
// X[b][m][e] = A[b][(e%8)*128 + m/8][(m%8)*16 + e/8] + PE[b*1024+(m%8)*128+e][m/8]
__global__ void k_buildX(const float* __restrict__ af2, float* __restrict__ X)
{
    int idx = blockIdx.x * blockDim.x + threadIdx.x;
    if (idx >= 4 * 1024 * 128) return;
    int e = idx & 127;
    int m = (idx >> 7) & 1023;
    int b = idx >> 17;
    int n = (e & 7) * 128 + (m >> 3);
    int d = (m & 7) * 16 + (e >> 3);
    float v = af2[((size_t)((b << 10) + n)) * 128 + d];
    int pos = (b << 10) + (m & 7) * 128 + e;
    int col = m >> 3;
    int j   = col >> 1;
    float div = __expf((float)(2 * j) * -0.0719557841f);  // -ln(10000)/128
    float ang = (float)pos * div;
    v += (col & 1) ? __cosf(ang) : __sinf(ang);
    X[idx] = v;
}

// Per-feature affine (BatchNorm1d eval): Y[i][e] = X[i][e]*g[e] + b[e]
__global__ void k_affine_lastdim(const float* __restrict__ Xin, float* __restrict__ Y,
                                 const float* __restrict__ g, const float* __restrict__ b)
{
    int idx = blockIdx.x * blockDim.x + threadIdx.x;
    if (idx >= 4 * 1024 * 128) return;
    int e = idx & 127;
    Y[idx] = Xin[idx] * g[e] + b[e];
}

// Row softmax over 1024 cols, then add identity (S is [4][1024][1024], in place).
__global__ __launch_bounds__(256) void k_softmax_eye(float* __restrict__ S)
{
    int row  = blockIdx.x;            // global row 0..4095
    int diag = row & 1023;
    float* p = S + (size_t)row * 1024;
    __shared__ float red[256];
    float m = -__builtin_inff();
    for (int c = threadIdx.x; c < 1024; c += 256) m = fmaxf(m, p[c]);
    red[threadIdx.x] = m; __syncthreads();
    for (int s = 128; s > 0; s >>= 1) {
        if (threadIdx.x < s) red[threadIdx.x] = fmaxf(red[threadIdx.x], red[threadIdx.x + s]);
        __syncthreads();
    }
    m = red[0]; __syncthreads();
    float sum = 0.f;
    for (int c = threadIdx.x; c < 1024; c += 256) sum += __expf(p[c] - m);
    red[threadIdx.x] = sum; __syncthreads();
    for (int s = 128; s > 0; s >>= 1) {
        if (threadIdx.x < s) red[threadIdx.x] += red[threadIdx.x + s];
        __syncthreads();
    }
    float inv = 1.f / red[0];
    for (int c = threadIdx.x; c < 1024; c += 256) {
        float v = __expf(p[c] - m) * inv;
        if (c == diag) v += 1.f;
        p[c] = v;
    }
}

// fc1 stage 1: partial[r][cz][c] = sum_{j in chunk cz} f[r][j] * w[j][c]
// f = [o1 | o2] per batch row; 64 chunks of 2048 rows; threads own column c
// so the 131072x128 weight matrix is read fully coalesced.
__global__ __launch_bounds__(128) void k_fc1_part(const float* __restrict__ o1,
                                                  const float* __restrict__ o2,
                                                  const float* __restrict__ w,
                                                  float* __restrict__ part)
{
    const int F = 65536;              // per-branch flat features per batch row
    int cz = blockIdx.x;              // 0..63
    int r  = blockIdx.y;              // 0..3
    int c  = threadIdx.x;             // 0..127
    const float* src = (cz < 32) ? (o1 + (size_t)r * F + (size_t)cz * 2048)
                                 : (o2 + (size_t)r * F + (size_t)(cz - 32) * 2048);
    const float* wp = w + (size_t)cz * 2048 * 128 + c;
    float sum = 0.f;
    for (int j = 0; j < 2048; ++j)
        sum += src[j] * wp[(size_t)j * 128];
    part[((size_t)r * 64 + cz) * 128 + c] = sum;
}

// fc1 stage 2: t1[r][c] = relu( sum_cz partial + bias[c] )
__global__ __launch_bounds__(128) void k_fc1_red(const float* __restrict__ part,
                                                 const float* __restrict__ bias,
                                                 float* __restrict__ t1)
{
    int r = blockIdx.x;
    int c = threadIdx.x;
    float sum = 0.f;
    for (int z = 0; z < 64; ++z)
        sum += part[((size_t)r * 64 + z) * 128 + c];
    t1[r * 128 + c] = fmaxf(sum + bias[c], 0.f);
}

// Tiny MLP tail: relu(t1@fc2+b) -> relu(@fc3+b) -> @fc4+b  => out[4][6]
__global__ __launch_bounds__(256) void k_head(const float* __restrict__ t1,
                                              const float* __restrict__ w2, const float* __restrict__ b2,
                                              const float* __restrict__ w3, const float* __restrict__ b3,
                                              const float* __restrict__ w4, const float* __restrict__ b4,
                                              float* __restrict__ out)
{
    __shared__ float s1[512], s2[512], s3[256];
    int tid = threadIdx.x;
    for (int i = tid; i < 512; i += 256) s1[i] = t1[i];
    __syncthreads();
    for (int i = tid; i < 512; i += 256) {
        int r = i >> 7, c = i & 127;
        float acc = b2[c];
        for (int k = 0; k < 128; ++k) acc += s1[r * 128 + k] * w2[k * 128 + c];
        s2[i] = fmaxf(acc, 0.f);
    }
    __syncthreads();
    if (tid < 256) {
        int r = tid >> 6, c = tid & 63;
        float acc = b3[c];
        for (int k = 0; k < 128; ++k) acc += s2[r * 128 + k] * w3[k * 64 + c];
        s3[tid] = fmaxf(acc, 0.f);
    }
    __syncthreads();
    if (tid < 24) {
        int r = tid / 6, c = tid % 6;
        float acc = b4[c];
        for (int k = 0; k < 64; ++k) acc += s3[r * 64 + k] * w4[k * 6 + c];
        out[tid] = acc;
    }
}

// ---------------------------------------------------------------------------
static inline void launch_gemm(hipStream_t st, const float* A, const float* B, float* C,
                               int M, int N, int K, int lda, int ldb, int ldc,
                               long long sA, long long sB, long long sC, int batch,
                               int transB, int mode,
                               const float* p0, const float* p1, const float* p2)
{
    int NT = (N + 15) / 16;
    dim3 grid(M / 16, (NT + 7) / 8, batch);
    if (transB)
        k_gemm_wmma_f32<1><<<grid, 256, 0, st>>>(A, B, C, M, N, K, lda, ldb, ldc,
                                                 sA, sB, sC, mode, p0, p1, p2);
    else
        k_gemm_wmma_f32<0><<<grid, 256, 0, st>>>(A, B, C, M, N, K, lda, ldb, ldc,
                                                 sA, sB, sC, mode, p0, p1, p2);
}

extern "C" void kernel_launch(void* const* d_in, const int* in_sizes, int n_in,
                              void* d_out, int out_size, void* d_ws, size_t ws_size,
                              hipStream_t stream)
{
    (void)in_sizes; (void)n_in; (void)out_size; (void)ws_size;
    const float* x_enc    = (const float*)d_in[0];
    const float* conv_w1  = (const float*)d_in[4];
    const float* bn1_g    = (const float*)d_in[5];
    const float* bn1_b    = (const float*)d_in[6];
    const float* conv_w2  = (const float*)d_in[7];
    const float* bn2_g    = (const float*)d_in[8];
    const float* bn2_b    = (const float*)d_in[9];
    const float* conv_w3  = (const float*)d_in[10];
    const float* bn3_g    = (const float*)d_in[11];
    const float* bn3_b    = (const float*)d_in[12];
    const float* map2_w   = (const float*)d_in[13];
    const float* map2_b   = (const float*)d_in[14];
    const float* map2_bng = (const float*)d_in[15];
    const float* map2_bnb = (const float*)d_in[16];
    const float* fc1_w    = (const float*)d_in[29];
    const float* fc1_b    = (const float*)d_in[30];
    const float* fc2_w    = (const float*)d_in[31];
    const float* fc2_b    = (const float*)d_in[32];
    const float* fc3_w    = (const float*)d_in[33];
    const float* fc3_b    = (const float*)d_in[34];
    const float* fc4_w    = (const float*)d_in[35];
    const float* fc4_b    = (const float*)d_in[36];

    float* W = (float*)d_ws;
    size_t off = 0;
    auto alloc = [&](size_t n) { float* p = W + off; off += n; return p; };
    float* BIG   = alloc(12582912);  // im2col2 (256*512*80) / im2col3 (256*1024*48); later ADJ/OPRE
    float* IM1   = alloc(294912);    // 256 * 8 * 144
    float* Y1C   = alloc(2113536);   // 256 * 64 * 129
    float* A1C   = alloc(1064960);   // 256 * 64 * 65
    float* Y2C   = alloc(2162688);   // 256 * 128 * 66
    float* A2C   = alloc(1114112);   // 256 * 128 * 34
    float* Y3C   = alloc(286720);    // 256 * 32 * 35
    float* AFEAT = alloc(2359296);   // 4096 * 576
    float* AF2   = alloc(524288);    // 4096 * 128
    float* Xb    = alloc(524288);    // X after PE
    float* XB    = alloc(524288);    // bn(X) per branch
    float* Hh    = alloc(524288);    // graph projection
    float* O1    = alloc(262144);    // 4096 * 64
    float* O2    = alloc(262144);
    float* PART  = alloc(32768);     // fc1 partials [4][64][128]
    float* T1    = alloc(512);       // fc1 output [4][128]
    float* ADJ   = BIG;              // 4 * 1024 * 1024 (conv scratch dead by then)
    float* OPRE  = BIG + 4194304;    // 4096 * 128

    const int CH = 256, NCH = 16;    // 4096 signals in 16 chunks
    for (int c = 0; c < NCH; ++c) {
        int s0 = c * CH;
        // conv1 (I=1, K=8, L 128 -> 129) as GEMM M=64 N=129 K=8 (B padded to 144)
        k_im2col_k8p4<<<dim3((8 * 144 + 255) / 256, 1, CH), 256, 0, stream>>>(
            x_enc + (size_t)s0 * 128, IM1, 1, 128, 144, 128LL, 8LL * 144);
        launch_gemm(stream, conv_w1, IM1, Y1C, 64, 129, 8, 8, 144, 129,
                    0, 8LL * 144, 64LL * 129, CH, 0, 5, bn1_g, bn1_b, nullptr);
        k_pool2p1<<<dim3((64 * 65 + 255) / 256, 1, CH), 256, 0, stream>>>(
            Y1C, A1C, 64, 129, 65, 64LL * 129, 64LL * 65);
        // conv2 (I=64, K=8, 65 -> 66) as GEMM M=128 N=66 K=512 (B padded to 80)
        k_im2col_k8p4<<<dim3((64 * 8 * 80 + 255) / 256, 1, CH), 256, 0, stream>>>(
            A1C, BIG, 64, 65, 80, 64LL * 65, 512LL * 80);
        launch_gemm(stream, conv_w2, BIG, Y2C, 128, 66, 512, 512, 80, 66,
                    0, 512LL * 80, 128LL * 66, CH, 0, 5, bn2_g, bn2_b, nullptr);
        k_pool2p1<<<dim3((128 * 34 + 255) / 256, 1, CH), 256, 0, stream>>>(
            Y2C, A2C, 128, 66, 34, 128LL * 66, 128LL * 34);
        // conv3 (I=128, K=8, 34 -> 35) as GEMM M=32 N=35 K=1024 (B padded to 48)
        k_im2col_k8p4<<<dim3((128 * 8 * 48 + 255) / 256, 1, CH), 256, 0, stream>>>(
            A2C, BIG, 128, 34, 48, 128LL * 34, 1024LL * 48);
        launch_gemm(stream, conv_w3, BIG, Y3C, 32, 35, 1024, 1024, 48, 35,
                    0, 1024LL * 48, 32LL * 35, CH, 0, 5, bn3_g, bn3_b, nullptr);
        // pool -> a_feat[s][o*18+u]
        k_pool2p1<<<dim3((32 * 18 + 255) / 256, 1, CH), 256, 0, stream>>>(
            Y3C, AFEAT + (size_t)s0 * 576, 32, 35, 18, 32LL * 35, 576LL);
    }

    // map2: [4096,576] @ [576,128], + bias then BN affine
    launch_gemm(stream, AFEAT, map2_w, AF2, 4096, 128, 576, 576, 128, 128,
                0, 0, 0, 1, 0, 1, map2_b, map2_bng, map2_bnb);

    // positional encoding + reshape permutation -> X [4][1024][128]
    k_buildX<<<2048, 256, 0, stream>>>(AF2, Xb);

    // two MPNN branches
    for (int br = 0; br < 2; ++br) {
        const float* gw  = (const float*)d_in[17 + br * 6];
        const float* gb  = (const float*)d_in[18 + br * 6];
        const float* bng = (const float*)d_in[19 + br * 6];
        const float* bnb = (const float*)d_in[20 + br * 6];
        const float* tw  = (const float*)d_in[21 + br * 6];
        const float* tb  = (const float*)d_in[22 + br * 6];
        float* Ob = (br == 0) ? O1 : O2;

        k_affine_lastdim<<<2048, 256, 0, stream>>>(Xb, XB, bng, bnb);
        // h = X @ gw + gb
        launch_gemm(stream, Xb, gw, Hh, 4096, 128, 128, 128, 128, 128,
                    0, 0, 0, 1, 0, 2, gb, nullptr, nullptr);
        // adj = leaky(h h^T - 1e8*eye)   (batched over 4)
        launch_gemm(stream, Hh, Hh, ADJ, 1024, 1024, 128, 128, 128, 1024,
                    1024LL * 128, 1024LL * 128, 1024LL * 1024, 4, 1, 4,
                    nullptr, nullptr, nullptr);
        // softmax rows + eye (in place)
        k_softmax_eye<<<4096, 256, 0, stream>>>(ADJ);
        // opre = adj @ XB   (batched over 4)
        launch_gemm(stream, ADJ, XB, OPRE, 1024, 128, 1024, 1024, 128, 128,
                    1024LL * 1024, 1024LL * 128, 1024LL * 128, 4, 0, 0,
                    nullptr, nullptr, nullptr);
        // o = leaky(opre @ tw + tb)
        launch_gemm(stream, OPRE, tw, Ob, 4096, 64, 128, 128, 64, 64,
                    0, 0, 0, 1, 0, 3, tb, nullptr, nullptr);
    }

    // fc1 over concat [o1|o2] (K=131072), then tiny head -> d_out [4][6]
    k_fc1_part<<<dim3(64, 4), 128, 0, stream>>>(O1, O2, fc1_w, PART);
    k_fc1_red<<<4, 128, 0, stream>>>(PART, fc1_b, T1);
    k_head<<<1, 256, 0, stream>>>(T1, fc2_w, fc2_b, fc3_w, fc3_b, fc4_w, fc4_b,
                                  (float*)d_out);
}